// CIF_90417651516460
// MI455X (gfx1250) — compile-verified
//
#include <hip/hip_runtime.h>
#include <math.h>

typedef __attribute__((ext_vector_type(2))) float v2f;
typedef __attribute__((ext_vector_type(8))) float v8f;

#define BB 16
#define SS 2048
#define DD 768
#define BS (BB*SS)   // 32768

// workspace layout (float offsets)
#define OFF_V     0        // 3*768 = 2304 : v_t planes, v[t*768+i]
#define OFF_C     2304     // 1           : scalar bias c
#define OFF_P     2320     // 3*32768     : P[n*BS + row] = x[row]·v_n
#define OFF_ALPHA 100624   // 32768
#define OFF_WA    133392   // 32768       : emit weight a1 / alpha
#define OFF_WB    166160   // 32768       : carry weight a2
#define OFF_CODE  198928   // 32768 ints  : out row (>=0) / -1 accumulate / -2 skip
// total 231696 floats = ~927 KB

// ---- Stage 1a: fold Linear into Conv:  v_t[i] = sum_o lin_w[o]*conv_w[o,i,t]
__global__ void k_fold_weights(const float* __restrict__ conv_w,
                               const float* __restrict__ conv_b,
                               const float* __restrict__ lin_w,
                               const float* __restrict__ lin_b,
                               float* __restrict__ ws) {
  int t = blockIdx.x * blockDim.x + threadIdx.x;
  if (t < DD * 3) {
    // flat t == i*3 + tt, conv_w stride over o is 2304 -> coalesced reads
    int i = t / 3, tt = t % 3;
    float acc = 0.f;
    for (int o = 0; o < DD; ++o) acc += lin_w[o] * conv_w[(size_t)o * (DD * 3) + t];
    ws[OFF_V + tt * DD + i] = acc;
  } else if (t == DD * 3) {
    float acc = lin_b[0];
    for (int o = 0; o < DD; ++o) acc += lin_w[o] * conv_b[o];
    ws[OFF_C] = acc;
  }
}

// ---- Stage 1b: P[n][row] = x[row] · v_n  via V_WMMA_F32_16X16X4_F32 (f32 matrix pipe)
// A tile 16x4: lanes 0-15 = rows M, VGPR0/1 = K+0/K+1 ; lanes 16-31 = K+2/K+3
// B tile 4x16: lane n = column (v_n), same K split as A
// C tile 16x16: VGPR j, lanes 0-15 -> M=j ; lanes 16-31 -> M=j+8
// Lanes n>=3 read a zero-filled 4th LDS plane so every lane issues an
// unconditional ds_load_b64 (no EXEC save/restore per iteration); only
// C columns 0..2 are ever stored.
__global__ void k_proj_wmma(const float* __restrict__ X,
                            const float* __restrict__ v_ws,
                            float* __restrict__ P) {
  __shared__ float vlds[4 * DD];
  for (int i = threadIdx.x; i < 4 * DD; i += blockDim.x)
    vlds[i] = (i < 3 * DD) ? v_ws[i] : 0.f;
  __syncthreads();

  const int lane = threadIdx.x & 31;
  const int wave = threadIdx.x >> 5;
  const int wgid = blockIdx.x * (blockDim.x >> 5) + wave;   // 0..2047
  const int r0   = wgid * 16;                               // 16 sequence rows per wave
  const int m    = lane & 15;
  const int hi   = lane >> 4;

  const int    np   = (m < 3) ? m : 3;                      // zero plane for unused cols
  const float* arow = X + (size_t)(r0 + m) * DD + 2 * hi;
  const float* bcol = vlds + np * DD + 2 * hi;

  v8f acc = {};
  for (int k0 = 0; k0 < DD; k0 += 4) {
    v2f a = *(const v2f*)(arow + k0);
    v2f b = *(const v2f*)(bcol + k0);
    acc = __builtin_amdgcn_wmma_f32_16x16x4_f32(false, a, false, b,
                                                (short)0, acc, false, false);
  }
  if (m < 3) {
    int base = r0 + hi * 8;
    #pragma unroll
    for (int j = 0; j < 8; ++j) P[(size_t)m * BS + base + j] = acc[j];
  }
}

// ---- Stage 1c: alpha[s] = sigmoid(P0[s-1] + P1[s] + P2[s+1] + c)
__global__ void k_alpha(const float* __restrict__ P, const float* __restrict__ cptr,
                        float* __restrict__ alphas) {
  int t = blockIdx.x * blockDim.x + threadIdx.x;
  if (t >= BS) return;
  int s = t & (SS - 1);
  float sum = P[BS + t] + cptr[0];
  if (s > 0)      sum += P[t - 1];
  if (s < SS - 1) sum += P[2 * BS + t + 1];
  alphas[t] = 1.f / (1.f + expf(-sum));
}

// ---- Stage 2a: sequential scalar CIF control stream per batch
__global__ void k_scan_ctrl(const float* __restrict__ alphas, const int* __restrict__ lens,
                            float* __restrict__ wa, float* __restrict__ wb,
                            int* __restrict__ code, float* __restrict__ lens_out) {
  int b = threadIdx.x;
  if (b >= BB) return;
  int len = lens[b];
  float aacc = 0.f;
  int cnt = 0;
  for (int s = 0; s < SS; ++s) {
    int idx = b * SS + s;
    float a = alphas[idx];
    bool valid = (s < len);
    float au = aacc + a;
    bool fire = (au >= 1.0f) && valid;
    if (fire) {
      float a1 = 1.0f - aacc;     // weight of h for the emitted segment
      float a2 = a - a1;          // leftover weight carried forward
      code[idx] = cnt; wa[idx] = a1; wb[idx] = a2;
      ++cnt; aacc = a2;
    } else if (valid) {
      code[idx] = -1; wa[idx] = a; wb[idx] = 0.f;
      aacc = au;
    } else {
      code[idx] = -2; wa[idx] = 0.f; wb[idx] = 0.f;
    }
  }
  lens_out[b] = (float)cnt;
}

// ---- Stage 2b: vector CIF apply; one thread per feature dim, hacc in register
#define CH 256
__global__ void k_cif_apply(const float* __restrict__ X,
                            const float* __restrict__ wa, const float* __restrict__ wb,
                            const int* __restrict__ code, float* __restrict__ out) {
  __shared__ float swa[CH];
  __shared__ float swb[CH];
  __shared__ int   scode[CH];

  const int chunks_per_b = DD / 256;                 // 3
  const int b = blockIdx.x / chunks_per_b;
  const int d = (blockIdx.x % chunks_per_b) * 256 + threadIdx.x;

  const float* xb = X   + (size_t)b * SS * DD + d;
  float*       ob = out + (size_t)b * SS * DD + d;

  float hacc = 0.f;
  for (int s0 = 0; s0 < SS; s0 += CH) {
    __syncthreads();
    int idx = b * SS + s0 + threadIdx.x;
    swa[threadIdx.x]   = wa[idx];
    swb[threadIdx.x]   = wb[idx];
    scode[threadIdx.x] = code[idx];
    __syncthreads();
    if (s0 + CH < SS) __builtin_prefetch(xb + (size_t)(s0 + CH) * DD, 0, 1);
    for (int s = 0; s < CH; ++s) {
      float h = xb[(size_t)(s0 + s) * DD];
      int cv = scode[s];
      if (cv >= 0) {                       // fire: emit, restart accumulator
        ob[(size_t)cv * DD] = hacc + swa[s] * h;
        hacc = swb[s] * h;
      } else if (cv == -1) {               // accumulate
        hacc += swa[s] * h;
      }                                    // cv == -2: past length, hold
    }
  }
}

extern "C" void kernel_launch(void* const* d_in, const int* in_sizes, int n_in,
                              void* d_out, int out_size, void* d_ws, size_t ws_size,
                              hipStream_t stream) {
  const float* X      = (const float*)d_in[0];   // (16,2048,768)
  const int*   lens   = (const int*)  d_in[1];   // (16,)
  const float* conv_w = (const float*)d_in[2];   // (768,768,3)
  const float* conv_b = (const float*)d_in[3];   // (768,)
  const float* lin_w  = (const float*)d_in[4];   // (1,768)
  const float* lin_b  = (const float*)d_in[5];   // (1,)
  float* ws  = (float*)d_ws;
  float* out = (float*)d_out;

  // output_c rows not fired must be zero; lens appended after B*S*D floats
  hipMemsetAsync(d_out, 0, (size_t)out_size * sizeof(float), stream);

  k_fold_weights<<<(DD * 3 + 1 + 255) / 256, 256, 0, stream>>>(conv_w, conv_b, lin_w, lin_b, ws);
  k_proj_wmma<<<BS / 16 / 8, 256, 0, stream>>>(X, ws + OFF_V, ws + OFF_P);
  k_alpha<<<(BS + 255) / 256, 256, 0, stream>>>(ws + OFF_P, ws + OFF_C, ws + OFF_ALPHA);
  k_scan_ctrl<<<1, 32, 0, stream>>>(ws + OFF_ALPHA, lens,
                                    ws + OFF_WA, ws + OFF_WB,
                                    (int*)(ws + OFF_CODE),
                                    out + (size_t)BB * SS * DD);
  k_cif_apply<<<BB * (DD / 256), 256, 0, stream>>>(X, ws + OFF_WA, ws + OFF_WB,
                                                   (const int*)(ws + OFF_CODE), out);
}